// GraphSAGE_53463752901314
// MI455X (gfx1250) — compile-verified
//
#include <hip/hip_runtime.h>
#include <hip/hip_bf16.h>

// GraphSAGE 3-layer forward for MI455X (gfx1250, wave32).
//  - Edge aggregation: wave-per-edge float4 gather + global_atomic_add_f32
//    (accumulator is 51MB -> L2-resident on the 192MB L2).
//  - Fused SAGE GEMM: out = act( [mean|self](16x256) @ [Wl;Wr](256xDOUT) + b )
//    using V_WMMA_F32_16X16X4_F32 with the A-tile staged in LDS per block.

typedef float v2f __attribute__((ext_vector_type(2)));
typedef float v8f __attribute__((ext_vector_type(8)));

#define N_CH 128              // input / hidden channels
#define LDS_STRIDE 260        // 16B-aligned rows, bank-conflict-free frags

// ---------------------------------------------------------------- degree ----
__global__ void sage_deg_kernel(const long long* __restrict__ dst,
                                float* __restrict__ deg, int nEdges) {
    int i = blockIdx.x * blockDim.x + threadIdx.x;
    if (i < nEdges) {
        atomicAdd(&deg[(int)dst[i]], 1.0f);
    }
}

__global__ void sage_rdeg_kernel(float* __restrict__ deg, int nNodes) {
    int i = blockIdx.x * blockDim.x + threadIdx.x;
    if (i < nNodes) {
        deg[i] = 1.0f / fmaxf(deg[i], 1.0f);   // in-place: deg -> 1/max(deg,1)
    }
}

// --------------------------------------------------------------- scatter ----
// One wave32 per edge: 128 channels = 32 lanes x float4. Coalesced 512B read
// of feat[src], 4 f32 atomics per lane into sum[dst] (L2-resident RMW).
__global__ void sage_scatter_kernel(const float* __restrict__ feat,
                                    const long long* __restrict__ src,
                                    const long long* __restrict__ dst,
                                    float* __restrict__ sum, int nEdges) {
    const int lane = threadIdx.x & 31;
    const int wid  = threadIdx.x >> 5;
    const long long e = (long long)blockIdx.x * (blockDim.x >> 5) + wid;
    if (e >= nEdges) return;

    const int s = (int)src[e];
    const int d = (int)dst[e];

    const float4 v = *(const float4*)(feat + (size_t)s * N_CH + lane * 4);
    float* p = sum + (size_t)d * N_CH + lane * 4;
    atomicAdd(p + 0, v.x);
    atomicAdd(p + 1, v.y);
    atomicAdd(p + 2, v.z);
    atomicAdd(p + 3, v.w);
}

// ------------------------------------------------------------- fused GEMM ---
// out[m, n] = act( sum_k mean[m,k]*Wl[k,n] + sum_k self[m,k]*Wr[k,n] + b[n] )
// Block = (DOUT/16) waves; block owns one 16-node M-tile, wave w owns N-tile w.
// A tile (16 x [mean(128)|self(128)]) staged in LDS once per block, mean
// scaling by rdeg fused into the staging loads.
template <int DOUT, bool RELU>
__global__ __launch_bounds__(32 * (DOUT / 16))
void sage_gemm_kernel(const float* __restrict__ S,    // aggregated sums [N,128]
                      const float* __restrict__ F,    // self features    [N,128]
                      const float* __restrict__ rdeg, // 1/max(deg,1)     [N]
                      const float* __restrict__ Wl,   // [128, DOUT]
                      const float* __restrict__ Wr,   // [128, DOUT]
                      const float* __restrict__ bias, // [DOUT]
                      float* __restrict__ out) {      // [N, DOUT]
    constexpr int NT   = DOUT / 16;
    constexpr int NTHR = 32 * NT;

    __shared__ float lds[16 * LDS_STRIDE];

    const int t     = threadIdx.x;
    const int lane  = t & 31;
    const int wave  = t >> 5;
    const int mbase = blockIdx.x * 16;

    // ---- cooperative A staging: 16 rows x 64 float4 (mean | self) ----
    for (int i = t; i < 16 * 64; i += NTHR) {
        const int row  = i >> 6;
        const int c4   = i & 63;
        const int node = mbase + row;
        float4 v;
        if (c4 < 32) {                       // mean part: sum * rdeg
            v = *(const float4*)(S + (size_t)node * N_CH + c4 * 4);
            const float r = rdeg[node];
            v.x *= r; v.y *= r; v.z *= r; v.w *= r;
        } else {                             // self part
            v = *(const float4*)(F + (size_t)node * N_CH + (c4 - 32) * 4);
        }
        *(float4*)(lds + row * LDS_STRIDE + c4 * 4) = v;   // ds_store_b128
    }
    __syncthreads();

    // ---- WMMA f32 16x16x4 fragment loop over K = 2*128 ----
    // A frag (16x4 f32): lane<16 -> {X[m][k0], X[m][k0+1]},
    //                    lane>=16 -> {X[m][k0+2], X[m][k0+3]},  m = lane&15.
    // B frag (4x16 f32): lane<16 -> {W[k0][n], W[k0+1][n]},
    //                    lane>=16 -> {W[k0+2][n], W[k0+3][n]},  n = n0+(lane&15).
    const int   n0     = wave * 16;
    const int   col    = n0 + (lane & 15);
    const int   koff   = (lane >> 4) * 2;
    const float* ldsRow = lds + (size_t)(lane & 15) * LDS_STRIDE;

    v8f acc = {};
    #pragma unroll 4
    for (int k0 = 0; k0 < N_CH; k0 += 4) {          // mean @ Wl
        const int kr = k0 + koff;
        v2f a = *(const v2f*)(ldsRow + kr);          // ds_load_b64
        v2f b;
        b.x = Wl[(size_t)kr * DOUT + col];
        b.y = Wl[(size_t)(kr + 1) * DOUT + col];
        acc = __builtin_amdgcn_wmma_f32_16x16x4_f32(
            false, a, false, b, (short)0, acc, false, false);
    }
    #pragma unroll 4
    for (int k0 = 0; k0 < N_CH; k0 += 4) {          // self @ Wr
        const int kr = k0 + koff;
        v2f a = *(const v2f*)(ldsRow + N_CH + kr);
        v2f b;
        b.x = Wr[(size_t)kr * DOUT + col];
        b.y = Wr[(size_t)(kr + 1) * DOUT + col];
        acc = __builtin_amdgcn_wmma_f32_16x16x4_f32(
            false, a, false, b, (short)0, acc, false, false);
    }

    // ---- epilogue: bias (+ReLU), C layout: VGPR v, row = v + 8*(lane>=16) ----
    const float bv     = bias[col];
    const int   rowoff = (lane >> 4) * 8;
    #pragma unroll
    for (int v = 0; v < 8; v++) {
        float r = acc[v] + bv;
        if (RELU) r = fmaxf(r, 0.0f);
        out[(size_t)(mbase + rowoff + v) * DOUT + col] = r;
    }
}

// ----------------------------------------------------------------- launch ---
extern "C" void kernel_launch(void* const* d_in, const int* in_sizes, int n_in,
                              void* d_out, int out_size, void* d_ws, size_t ws_size,
                              hipStream_t stream) {
    const float*     x   = (const float*)d_in[0];
    const long long* ei  = (const long long*)d_in[1];   // int64 [2, E]
    const float*     Wl0 = (const float*)d_in[2];
    const float*     Wr0 = (const float*)d_in[3];
    const float*     b0  = (const float*)d_in[4];
    const float*     Wl1 = (const float*)d_in[5];
    const float*     Wr1 = (const float*)d_in[6];
    const float*     b1  = (const float*)d_in[7];
    const float*     Wl2 = (const float*)d_in[8];
    const float*     Wr2 = (const float*)d_in[9];
    const float*     b2  = (const float*)d_in[10];
    float*           out = (float*)d_out;

    const int N = in_sizes[0] / N_CH;     // 100000
    const int E = in_sizes[1] / 2;        // 1600000

    const long long* src = ei;
    const long long* dst = ei + E;

    // workspace layout: S (sum, reused) | H1 | H2 | RDEG
    const size_t NF = (size_t)N * N_CH;
    float* S    = (float*)d_ws;
    float* H1   = S + NF;
    float* H2   = H1 + NF;
    float* RDEG = H2 + NF;

    const int mtiles = N / 16;            // 100000/16 = 6250 exact

    // degree (dst-only -> identical for all layers, computed once)
    hipMemsetAsync(RDEG, 0, (size_t)N * sizeof(float), stream);
    sage_deg_kernel<<<(E + 255) / 256, 256, 0, stream>>>(dst, RDEG, E);
    sage_rdeg_kernel<<<(N + 255) / 256, 256, 0, stream>>>(RDEG, N);

    // ---- layer 0: x -> H1 (ReLU) ----
    hipMemsetAsync(S, 0, NF * sizeof(float), stream);
    sage_scatter_kernel<<<(E + 7) / 8, 256, 0, stream>>>(x, src, dst, S, E);
    sage_gemm_kernel<128, true><<<mtiles, 256, 0, stream>>>(
        S, x, RDEG, Wl0, Wr0, b0, H1);

    // ---- layer 1: H1 -> H2 (ReLU) ----
    hipMemsetAsync(S, 0, NF * sizeof(float), stream);
    sage_scatter_kernel<<<(E + 7) / 8, 256, 0, stream>>>(H1, src, dst, S, E);
    sage_gemm_kernel<128, true><<<mtiles, 256, 0, stream>>>(
        S, H1, RDEG, Wl1, Wr1, b1, H2);

    // ---- layer 2: H2 -> out (no activation, DOUT=64) ----
    hipMemsetAsync(S, 0, NF * sizeof(float), stream);
    sage_scatter_kernel<<<(E + 7) / 8, 256, 0, stream>>>(H2, src, dst, S, E);
    sage_gemm_kernel<64, false><<<mtiles, 128, 0, stream>>>(
        S, H2, RDEG, Wl2, Wr2, b2, out);
}